// GPR_58746562675312
// MI455X (gfx1250) — compile-verified
//
#include <hip/hip_runtime.h>
#include <math.h>

typedef __attribute__((ext_vector_type(2))) float v2f;
typedef __attribute__((ext_vector_type(4))) float v4f;
typedef __attribute__((ext_vector_type(8))) float v8f;

#define X_DIM 20000
#define Y_DIM 6000
#define X_TILES (X_DIM / 16)   // 1250
#define Y_TILES (Y_DIM / 16)   // 375
#define D2R 0.017453292519943295f
// sin^2(0.25 / 12.756): hav < this  <=>  d < 0.25
#define HAV_THRESH 3.8405686e-4f
// EDIAM * asin-series coefficients: d = z*(C0 + hav*(C1 + hav*C2)), z=sqrt(hav)
#define WC0 12.756f        // EDIAM
#define WC1 2.1260f        // EDIAM/6
#define WC2 0.9567f        // EDIAM*3/40

// Coefficient tables are stored GROUPED BY LANE-HALF so the WMMA operand for
// lane-half h is one contiguous 16-byte chunk:
//   slots 0..3 = {K0, K1, K4, K5}   (h = 0 needs K 2h,2h+1,4+2h,5+2h = 0,1,4,5)
//   slots 4..7 = {K2, K3, K6, K7}   (h = 1 needs 2,3,6,7)

// ---- Precompute x-side rank-6 coefficients (+ compact x-value array) ----
__global__ void precompute_x_coeffs(const float* __restrict__ x,
                                    const float* __restrict__ x_l,
                                    float* __restrict__ Ax,
                                    float* __restrict__ Xv) {
    int i = blockIdx.x * blockDim.x + threadIdx.x;
    if (i >= X_DIM) return;
    float lon = x_l[2 * i] * D2R;
    float lat = x_l[2 * i + 1] * D2R;
    float sa, ca, sp, cp;
    sincosf(lat, &sa, &ca);
    sincosf(lon, &sp, &cp);
    float k0 = sa * sa;
    float k1 = -2.0f * sa * ca;
    float k2 = ca * ca;
    float k3 = ca * sp * sp;
    float k4 = -2.0f * ca * sp * cp;
    float k5 = ca * cp * cp;
    float* o = Ax + (size_t)i * 8;
    o[0] = k0; o[1] = k1; o[2] = k4; o[3] = k5;   // h=0 group: K0,K1,K4,K5
    o[4] = k2; o[5] = k3; o[6] = 0.f; o[7] = 0.f; // h=1 group: K2,K3,K6,K7
    Xv[i] = x[i];
}

// ---- Precompute y-side rank-6 coefficients ----
__global__ void precompute_y_coeffs(const float* __restrict__ y_l,
                                    float* __restrict__ By) {
    int i = blockIdx.x * blockDim.x + threadIdx.x;
    if (i >= Y_DIM) return;
    float lon = y_l[2 * i] * D2R;
    float lat = y_l[2 * i + 1] * D2R;
    float sb, cb, sp, cp;
    sincosf(lat, &sb, &cb);
    sincosf(lon, &sp, &cp);
    float k0 = cb * cb;
    float k1 = sb * cb;
    float k2 = sb * sb;
    float k3 = cb * cp * cp;
    float k4 = cb * cp * sp;
    float k5 = cb * sp * sp;
    float* o = By + (size_t)i * 8;
    o[0] = k0; o[1] = k1; o[2] = k4; o[3] = k5;   // h=0 group
    o[4] = k2; o[5] = k3; o[6] = 0.f; o[7] = 0.f; // h=1 group
}

// ---- Main kernel: hav tiles via V_WMMA_F32_16X16X4_F32 (K=6 padded to 8) ----
// Block: 128 threads = 4 wave32. Block b owns y rows [16b, 16b+16).
__global__ __launch_bounds__(128) void hav_weighted_mean(
        const float* __restrict__ Ax,
        const float* __restrict__ Xv,
        const float* __restrict__ By,
        float* __restrict__ out) {
    __shared__ float red_s[4][16];
    __shared__ float red_m[4][16];

    const int tid  = threadIdx.x;
    const int wave = tid >> 5;
    const int lane = tid & 31;
    const int h    = lane >> 4;   // lane half (selects K group per ISA layout)
    const int n    = lane & 15;   // row (A) / column (B) index within tile

    // A-matrix operand (y coefficients): one b128 per lane, loaded once.
    v4f aq = *(const v4f*)(By + (size_t)(blockIdx.x * 16 + n) * 8 + h * 4);
    v2f a0, a1;
    a0.x = aq.x; a0.y = aq.y;   // K = 2h, 2h+1
    a1.x = aq.z; a1.y = aq.w;   // K = 4+2h, 5+2h

    // Packed accumulators: sacc2[p] covers C rows 2p, 2p+1.
    v2f sacc2[4] = {{0.f, 0.f}, {0.f, 0.f}, {0.f, 0.f}, {0.f, 0.f}};
    v2f macc2[4] = {{0.f, 0.f}, {0.f, 0.f}, {0.f, 0.f}, {0.f, 0.f}};

    const v2f wc0v = {WC0, WC0};
    const v2f wc1v = {WC1, WC1};
    const v2f wc2v = {WC2, WC2};

    // Pointer-bump addressing: each wave strides 4 tiles = 64 x-points.
    const float* axp = Ax + ((size_t)wave * 16 + n) * 8 + h * 4;
    const float* xvp = Xv + wave * 16 + n;

    for (int t = wave; t < X_TILES; t += 4) {
        v4f bq = *(const v4f*)axp;   // 1x global_load_b128
        float xv = *xvp;             // 1x global_load_b32
        axp += 4 * 16 * 8;
        xvp += 4 * 16;
        v2f b0, b1;
        b0.x = bq.x; b0.y = bq.y;
        b1.x = bq.z; b1.y = bq.w;

        v8f c = {};
        c = __builtin_amdgcn_wmma_f32_16x16x4_f32(
                false, a0, false, b0, (short)0, c, false, false);
        c = __builtin_amdgcn_wmma_f32_16x16x4_f32(
                false, a1, false, b1, (short)0, c, false, false);

        // c[r] = hav for y-row (r + 8h), x-col n of this tile.
        // Passing pairs have sqrt(hav) < 0.0196 => 3-term asin series is
        // exact to fp32; raw v_sqrt_f32 (1 ulp) is plenty accurate here.
        // Explicitly packed epilogue: scalar v_med3/v_sqrt (no packed form),
        // everything else as v2f -> v_pk_fma/v_pk_mul/v_pk_add.
        const v2f xv2 = {xv, xv};
        #pragma unroll
        for (int p = 0; p < 4; ++p) {
            v2f hav2;
            hav2.x = __builtin_amdgcn_fmed3f(c[2 * p],     0.0f, 1e30f);
            hav2.y = __builtin_amdgcn_fmed3f(c[2 * p + 1], 0.0f, 1e30f);
            v2f z2;
            z2.x = __builtin_amdgcn_sqrtf(hav2.x);
            z2.y = __builtin_amdgcn_sqrtf(hav2.y);
            v2f poly2 = __builtin_elementwise_fma(
                hav2, __builtin_elementwise_fma(hav2, wc2v, wc1v), wc0v);
            v2f d2 = z2 * poly2;                        // v_pk_mul_f32
            v2f w2;
            w2.x = (hav2.x < HAV_THRESH) ? d2.x : 0.0f;
            w2.y = (hav2.y < HAV_THRESH) ? d2.y : 0.0f;
            sacc2[p] = sacc2[p] + w2;                   // v_pk_add_f32
            macc2[p] = __builtin_elementwise_fma(w2, xv2, macc2[p]); // pk_fma
        }
    }

    // Reduce across the 16 columns (lanes within each half).
    #pragma unroll
    for (int r = 0; r < 8; ++r) {
        float s = (r & 1) ? sacc2[r >> 1].y : sacc2[r >> 1].x;
        float m = (r & 1) ? macc2[r >> 1].y : macc2[r >> 1].x;
        #pragma unroll
        for (int off = 8; off >= 1; off >>= 1) {
            s += __shfl_xor(s, off, 32);
            m += __shfl_xor(m, off, 32);
        }
        if (n == 0) {
            red_s[wave][r + 8 * h] = s;
            red_m[wave][r + 8 * h] = m;
        }
    }
    __syncthreads();

    // Combine the 4 waves' partial x-ranges and write outputs.
    if (tid < 16) {
        float s = red_s[0][tid] + red_s[1][tid] + red_s[2][tid] + red_s[3][tid];
        float m = red_m[0][tid] + red_m[1][tid] + red_m[2][tid] + red_m[3][tid];
        int row = blockIdx.x * 16 + tid;
        out[row]         = (s > 0.0f) ? (m / fmaxf(s, 1e-12f)) : 0.0f;
        out[Y_DIM + row] = 0.001f;   // v_diag
    }
}

extern "C" void kernel_launch(void* const* d_in, const int* in_sizes, int n_in,
                              void* d_out, int out_size, void* d_ws, size_t ws_size,
                              hipStream_t stream) {
    (void)in_sizes; (void)n_in; (void)out_size; (void)ws_size;
    const float* x   = (const float*)d_in[0];   // (20000,1)
    const float* x_l = (const float*)d_in[1];   // (20000,2) lon,lat
    const float* y_l = (const float*)d_in[2];   // (6000,2)  lon,lat
    float* out = (float*)d_out;                 // m (6000) ++ v_diag (6000)

    float* Ax = (float*)d_ws;                   // 20000*8 f32 = 640 KB
    float* Xv = Ax + (size_t)X_DIM * 8;         // 20000   f32 =  80 KB
    float* By = Xv + X_DIM;                     //  6000*8 f32 = 192 KB

    precompute_x_coeffs<<<(X_DIM + 255) / 256, 256, 0, stream>>>(x, x_l, Ax, Xv);
    precompute_y_coeffs<<<(Y_DIM + 255) / 256, 256, 0, stream>>>(y_l, By);
    hav_weighted_mean<<<Y_TILES, 128, 0, stream>>>(Ax, Xv, By, out);
}